// SpecialLoss_3710851744335
// MI455X (gfx1250) — compile-verified
//
#include <hip/hip_runtime.h>
#include <hip/hip_bf16.h>
#include <stdint.h>

// Problem shape (fixed by the reference).
#define TB 4
#define TN 256
#define TH 256
#define TW 256
#define THW (TH * TW)        // 65536
#define CN 32                // channels per TDM chunk (32 KB tile)
#define NCHUNK (TN / CN)     // 8 chunks

typedef unsigned int u32;
typedef u32 u32x4 __attribute__((ext_vector_type(4)));
typedef int i32x4 __attribute__((ext_vector_type(4)));
typedef int i32x8 __attribute__((ext_vector_type(8)));

// ---------------------------------------------------------------------------
// TDM: DMA one 2D tile (CN rows of TW int32, row stride THW elements) from
// global memory into LDS at byte offset lds_off. D# layout per CDNA5 ISA §8.
// ---------------------------------------------------------------------------
__device__ __forceinline__ void tdm_load_tile(const int* gptr, u32 lds_off) {
  unsigned long long ga = (unsigned long long)(uintptr_t)gptr;
  u32x4 g0;
  g0.x = 1u;                                   // count=1, is_restore=0, gather off
  g0.y = lds_off;                              // lds_addr (bytes, wave-relative)
  g0.z = (u32)ga;                              // global_addr[31:0]
  g0.w = (u32)((ga >> 32) & 0x01FFFFFFu)       // global_addr[56:32]
         | 0x80000000u;                        // type = 2 ("image")
  i32x8 g1;
  g1[0] = 0x00020000;                          // workgroup_mask=0, data_size=2 (4B)
  g1[1] = (int)((u32)TW << 16);                // tensor_dim0 = 256 (lo16 in [31:16])
  g1[2] = (int)((u32)CN << 16);                // dim0 hi=0 | tensor_dim1 = CN
  g1[3] = (int)((u32)TW << 16);                // dim1 hi=0 | tile_dim0 = 256
  g1[4] = CN;                                  // tile_dim1 = CN | tile_dim2 = 0 (2D)
  g1[5] = THW;                                 // tensor_dim0_stride = 65536 (lo32)
  g1[6] = 0;                                   // stride0 hi | stride1 lo
  g1[7] = 0;
  i32x4 z4 = {0, 0, 0, 0};
#if __clang_major__ >= 23
  i32x8 z8 = {0, 0, 0, 0, 0, 0, 0, 0};
  __builtin_amdgcn_tensor_load_to_lds(g0, g1, z4, z4, z8, 0);
#else
  __builtin_amdgcn_tensor_load_to_lds(g0, g1, z4, z4, 0);
#endif
}

// ---------------------------------------------------------------------------
// Kernel 1: compact the one-hot center tensor into flat HW indices.
// B128 streaming loads; exactly one write per (b,n) channel.
// ---------------------------------------------------------------------------
__global__ void centers_kernel(const int* __restrict__ tc,
                               int* __restrict__ centers, int n4) {
  size_t i = (size_t)blockIdx.x * blockDim.x + threadIdx.x;
  size_t stride = (size_t)gridDim.x * blockDim.x;
  const int4* p = (const int4*)tc;
  for (; i < (size_t)n4; i += stride) {
    int4 v = p[i];
    if (v.x | v.y | v.z | v.w) {
      size_t e = i * 4;
      if (v.x) centers[(e + 0) >> 16] = (int)((e + 0) & 65535);
      if (v.y) centers[(e + 1) >> 16] = (int)((e + 1) & 65535);
      if (v.z) centers[(e + 2) >> 16] = (int)((e + 2) & 65535);
      if (v.w) centers[(e + 3) >> 16] = (int)((e + 3) & 65535);
    }
  }
}

// ---------------------------------------------------------------------------
// Kernel 2: one block per (b,h) row, one thread per w. Targets streamed via
// double-buffered TDM tiles through LDS. Channel metadata packed into one
// LDS word, decoded on the SALU pipe (uniform across the block).
// ---------------------------------------------------------------------------
__global__ void __launch_bounds__(256)
main_kernel(const float* __restrict__ pred, const int* __restrict__ targets,
            const int* __restrict__ pad, const int* __restrict__ centers,
            float* __restrict__ partials) {
  __shared__ int tileA[CN * TW];   // 32 KB
  __shared__ int tileB[CN * TW];   // 32 KB
  __shared__ int pack_s[TN];       // cc | cr<<8 | msk<<16 (uniform per block)
  __shared__ float red_s[256];

  const int tid = threadIdx.x;
  const int blk = blockIdx.x;
  const int b = blk >> 8;
  const int h = blk & 255;
  const int w = tid;

  // LDS byte offsets of the two tile buffers (flat LDS addr = low 32 bits).
  const u32 offA = (u32)(uintptr_t)(void*)&tileA[0];
  const u32 offB = (u32)(uintptr_t)(void*)&tileB[0];

  // Preload per-b channel metadata: flat center idx (cr<<8|cc) + !pad bit.
  {
    const int v = centers[b * TN + tid] & 0xFFFF;
    const int m = pad[b * TN + tid] ? 0 : 1;
    pack_s[tid] = v | (m << 16);
  }

  const int* tbase = targets + ((size_t)b * TN * THW + (size_t)h * TW);

  if (tid == 0) tdm_load_tile(tbase, offA);  // prologue: chunk 0 -> tileA

  // Hoist the prediction load so it retires under the streaming phase.
  const float x = pred[(size_t)b * THW + (size_t)h * TW + w];

  float pi = 0.0f;   // sum_n importance contribution
  float tsf = 0.0f;  // sum_n targets (exact in fp32, <= 256)

  for (int c = 0; c < NCHUNK; ++c) {
    if (tid == 0) {
      if (c + 1 < NCHUNK) {
        tdm_load_tile(tbase + (size_t)(c + 1) * CN * THW,
                      ((c + 1) & 1) ? offB : offA);
        __builtin_amdgcn_s_wait_tensorcnt(1);  // chunk c landed (in-order)
      } else {
        __builtin_amdgcn_s_wait_tensorcnt(0);
      }
    }
    __syncthreads();  // chunk c visible to all waves (also covers metadata)

    const int* tl = (c & 1) ? tileB : tileA;
#pragma unroll
    for (int j = 0; j < CN; ++j) {
      const int n = c * CN + j;
      const int t = tl[j * TW + tid];            // 1 DS load per element
      const float tf = (float)t;
      tsf += tf;
      // Uniform metadata -> SGPRs: decode + (h-cr)^2 ride the SALU pipe,
      // and padded channels (~half) skip the reciprocal path via a scalar
      // branch (they only contribute to the target sum).
      const int pk = __builtin_amdgcn_readfirstlane(pack_s[n]);
      if (pk >> 16) {
        const int cr = (pk >> 8) & 255;          // SALU
        const int cc = pk & 255;                 // SALU
        const int dr2 = (h - cr) * (h - cr);     // SALU (uniform)
        const int dc = w - cc;                   // VALU
        const float d2 = (float)(dr2 + dc * dc) + 1e-6f;
        pi += tf * __builtin_amdgcn_rcpf(d2);    // trans pipe co-executes
      }
    }
    __syncthreads();  // everyone done reading this buffer before reuse
  }

  // importance_coeff = (~tsum) * 0.5 = (-tsum - 1) * 0.5
  const float z = tsf;
  const float pitot = pi + (-z - 1.0f) * 0.5f;
  const float bce = fmaxf(x, 0.0f) - x * z + __logf(1.0f + __expf(-fabsf(x)));
  const float val = bce * pitot;

  // Deterministic tree reduction.
  red_s[tid] = val;
  __syncthreads();
  for (int off = 128; off > 0; off >>= 1) {
    if (tid < off) red_s[tid] += red_s[tid + off];
    __syncthreads();
  }
  if (tid == 0) partials[blk] = red_s[0];
}

// ---------------------------------------------------------------------------
// Kernel 3: deterministic final reduction of 1024 partials -> mean.
// ---------------------------------------------------------------------------
__global__ void reduce_kernel(const float* __restrict__ partials,
                              float* __restrict__ out, int npart, float scale) {
  __shared__ float red[256];
  const int tid = threadIdx.x;
  float s = 0.0f;
  for (int i = tid; i < npart; i += 256) s += partials[i];
  red[tid] = s;
  __syncthreads();
  for (int off = 128; off > 0; off >>= 1) {
    if (tid < off) red[tid] += red[tid + off];
    __syncthreads();
  }
  if (tid == 0) out[0] = red[0] * scale;
}

// ---------------------------------------------------------------------------
extern "C" void kernel_launch(void* const* d_in, const int* in_sizes, int n_in,
                              void* d_out, int out_size, void* d_ws,
                              size_t ws_size, hipStream_t stream) {
  (void)in_sizes; (void)n_in; (void)out_size; (void)ws_size;
  const float* pred = (const float*)d_in[0];
  const int* tcent  = (const int*)d_in[1];
  const int* targ   = (const int*)d_in[2];
  const int* pad    = (const int*)d_in[3];
  float* out = (float*)d_out;

  int* centers = (int*)d_ws;                                 // B*N ints = 4 KB
  float* partials = (float*)((char*)d_ws + 8 * 1024);        // 1024 floats

  const int n4 = (TB * TN * TH * TW) / 4;  // int4 count of one-hot tensor
  centers_kernel<<<8192, 256, 0, stream>>>(tcent, centers, n4);
  main_kernel<<<TB * TH, 256, 0, stream>>>(pred, targ, pad, centers, partials);
  reduce_kernel<<<1, 256, 0, stream>>>(partials, out, TB * TH,
                                       1.0f / (float)(TB * TH * TW));
}